// ChamferLossEF_82008105549924
// MI455X (gfx1250) — compile-verified
//
#include <hip/hip_runtime.h>
#include <hip/hip_bf16.h>

typedef __attribute__((ext_vector_type(2))) float v2f;
typedef __attribute__((ext_vector_type(8))) float v8f;

#define V_PTS   16384
#define NTILES  (V_PTS / 16)      // 1024 16-wide column tiles per direction
#define GROUP   4                 // column tiles per software-pipeline stage
#define NGROUPS (NTILES / GROUP)  // 256
#define PAD_PTS (GROUP * 16)      // 64 padded reference entries for prefetch overrun
#define PACKED_FLOATS ((size_t)(V_PTS + PAD_PTS) * 4)

// Pack reference points into WMMA B-operand form: (-2x, -2y, -2z, x^2+y^2+z^2).
// Score = A(16x4) * B(4x16) with A-row = (qx,qy,qz,1)  ->  -2 q.r + ||r||^2,
// which has the same argmin over columns as the true squared distance.
__global__ __launch_bounds__(256)
void chamfer_prep_kernel(const float* __restrict__ pv,
                         const float* __restrict__ tv,
                         float* __restrict__ packedP,
                         float* __restrict__ packedT,
                         float* __restrict__ out) {
    int i = blockIdx.x * blockDim.x + threadIdx.x;
    if (i == 0) out[0] = 0.0f;           // d_out is poisoned; reset every call
    if (i < V_PTS) {
        float x = pv[3 * i + 0], y = pv[3 * i + 1], z = pv[3 * i + 2];
        packedP[4 * i + 0] = -2.0f * x;
        packedP[4 * i + 1] = -2.0f * y;
        packedP[4 * i + 2] = -2.0f * z;
        packedP[4 * i + 3] = x * x + y * y + z * z;
        x = tv[3 * i + 0]; y = tv[3 * i + 1]; z = tv[3 * i + 2];
        packedT[4 * i + 0] = -2.0f * x;
        packedT[4 * i + 1] = -2.0f * y;
        packedT[4 * i + 2] = -2.0f * z;
        packedT[4 * i + 3] = x * x + y * y + z * z;
    } else if (i < V_PTS + PAD_PTS) {    // zero the prefetch-overrun pad
        packedP[4 * i + 0] = 0.0f; packedP[4 * i + 1] = 0.0f;
        packedP[4 * i + 2] = 0.0f; packedP[4 * i + 3] = 0.0f;
        packedT[4 * i + 0] = 0.0f; packedT[4 * i + 1] = 0.0f;
        packedT[4 * i + 2] = 0.0f; packedT[4 * i + 3] = 0.0f;
    }
}

// One wave32 per 16-row query tile. 1024 tiles per direction, 2 directions.
// Software-pipelined: prefetch the next GROUP of column tiles (4 b64 loads in
// flight) while running GROUP wmma+argmin stages on the current group.
__global__ __launch_bounds__(256)
void chamfer_nn_kernel(const float* __restrict__ pv,
                       const float* __restrict__ tv,
                       const float* __restrict__ packedP,
                       const float* __restrict__ packedT,
                       int* __restrict__ t2p,
                       int* __restrict__ p2t) {
    const int lane    = threadIdx.x & 31;
    const int wid     = blockIdx.x * (blockDim.x >> 5) + (threadIdx.x >> 5);
    const int dir     = wid >> 10;              // 0: q=trg vs pred refs, 1: q=pred vs trg refs
    const int rowTile = wid & (NTILES - 1);

    const float* q      = dir ? pv      : tv;
    const float* pr     = dir ? packedT : packedP;
    int*         outIdx = dir ? p2t     : t2p;

    const int half = lane >> 4;                 // A layout: lanes 0-15 hold K=0,1; 16-31 hold K=2,3
    const int m    = lane & 15;
    const int row  = rowTile * 16 + m;

    v2f a;
    if (half == 0) { a.x = q[3 * row + 0]; a.y = q[3 * row + 1]; }
    else           { a.x = q[3 * row + 2]; a.y = 1.0f; }

    float best[8];
    int   bidx[8];
#pragma unroll
    for (int i = 0; i < 8; ++i) { best[i] = 3.4e38f; bidx[i] = 0; }

    const v2f* __restrict__ bbase = (const v2f*)pr;   // packed [V+pad][4] viewed as [.][2] x v2f
    const int  laneoff = m * 2 + half;                // this lane's v2f slot within a tile

    v2f b[GROUP], bn[GROUP];
#pragma unroll
    for (int j = 0; j < GROUP; ++j) b[j] = bbase[j * 32 + laneoff];   // 16 pts * 2 v2f = 32/tile

    for (int g = 0; g < NGROUPS; ++g) {
        // prefetch next group (g==NGROUPS-1 reads the zeroed pad; never used in a wmma)
        const int nslot = (g + 1) * GROUP * 32;
#pragma unroll
        for (int j = 0; j < GROUP; ++j) bn[j] = bbase[nslot + j * 32 + laneoff];

#pragma unroll
        for (int j = 0; j < GROUP; ++j) {
            v8f c = {};
            c = __builtin_amdgcn_wmma_f32_16x16x4_f32(
                    /*neg_a=*/false, a, /*neg_b=*/false, b[j],
                    /*c_mod=*/(short)0, c, /*reuse_a=*/false, /*reuse_b=*/false);
            const int col = (g * GROUP + j) * 16 + m;  // this lane's column
            // C layout: VGPR i -> row (i + 8*half)
#pragma unroll
            for (int i = 0; i < 8; ++i) {
                if (c[i] < best[i]) { best[i] = c[i]; bidx[i] = col; }
            }
        }
#pragma unroll
        for (int j = 0; j < GROUP; ++j) b[j] = bn[j];
    }

    // min+argmin reduction across the 16 lanes of each half-wave
    // (xor masks 1..8 never cross the lane-16 boundary)
#pragma unroll
    for (int mask = 1; mask < 16; mask <<= 1) {
#pragma unroll
        for (int i = 0; i < 8; ++i) {
            float ov = __shfl_xor(best[i], mask, 32);
            int   oi = __shfl_xor(bidx[i], mask, 32);
            if (ov < best[i] || (ov == best[i] && oi < bidx[i])) {
                best[i] = ov; bidx[i] = oi;
            }
        }
    }
    if (m == 0) {                                  // lanes 0 (rows 0-7) and 16 (rows 8-15)
        const int rbase = rowTile * 16 + half * 8;
#pragma unroll
        for (int i = 0; i < 8; ++i) outIdx[rbase + i] = bidx[i];
    }
}

__global__ __launch_bounds__(256)
void chamfer_loss_kernel(const float* __restrict__ pe,
                         const float* __restrict__ te,
                         const int* __restrict__ t2p,
                         const int* __restrict__ p2t,
                         float* __restrict__ out) {
    __shared__ float red[8];
    int i = blockIdx.x * blockDim.x + threadIdx.x;
    float acc = 0.0f;
    if (i < V_PTS) {
        int j1 = t2p[i];   // trg_e[i] vs pred_e[j1]
        int j2 = p2t[i];   // pred_e[i] vs trg_e[j2]
#pragma unroll
        for (int c = 0; c < 3; ++c) {
            float d1 = te[3 * i + c] - pe[3 * j1 + c];
            float d2 = pe[3 * i + c] - te[3 * j2 + c];
            acc += d1 * d1 + d2 * d2;
        }
    }
#pragma unroll
    for (int mask = 16; mask >= 1; mask >>= 1) acc += __shfl_xor(acc, mask, 32);
    const int lane = threadIdx.x & 31;
    const int w    = threadIdx.x >> 5;
    if (lane == 0) red[w] = acc;
    __syncthreads();
    if (threadIdx.x == 0) {
        float s = 0.0f;
#pragma unroll
        for (int k = 0; k < 8; ++k) s += red[k];
        // each of the two means divides by V*3
        atomicAdd(out, s * (1.0f / (3.0f * (float)V_PTS)));
    }
}

extern "C" void kernel_launch(void* const* d_in, const int* in_sizes, int n_in,
                              void* d_out, int out_size, void* d_ws, size_t ws_size,
                              hipStream_t stream) {
    const float* pv = (const float*)d_in[0];   // pred_vertices [1,V,3]
    const float* tv = (const float*)d_in[1];   // trg_vertices  [1,V,3]
    const float* pe = (const float*)d_in[2];   // pred_e        [1,V,3]
    const float* te = (const float*)d_in[3];   // trg_e         [1,V,3]
    float* out = (float*)d_out;

    char* ws = (char*)d_ws;
    float* packedP = (float*)(ws);
    float* packedT = (float*)(ws + PACKED_FLOATS * sizeof(float));
    int*   t2p     = (int*)  (ws + 2 * PACKED_FLOATS * sizeof(float));
    int*   p2t     = (int*)  (ws + 2 * PACKED_FLOATS * sizeof(float) + (size_t)V_PTS * sizeof(int));

    chamfer_prep_kernel<<<(V_PTS + PAD_PTS + 255) / 256, 256, 0, stream>>>(
        pv, tv, packedP, packedT, out);

    // 2 directions * 1024 row tiles = 2048 waves; 8 waves per 256-thread block
    chamfer_nn_kernel<<<(2 * NTILES) / 8, 256, 0, stream>>>(pv, tv, packedP, packedT, t2p, p2t);

    chamfer_loss_kernel<<<(V_PTS + 255) / 256, 256, 0, stream>>>(pe, te, t2p, p2t, out);
}